// BasicFlow_4861902979633
// MI455X (gfx1250) — compile-verified
//
#include <hip/hip_runtime.h>
#include <math.h>
#include <stdint.h>

// ---------------------------------------------------------------------------
// Problem constants (from reference): B=2, C=256, H=W=256, PART=8
// ---------------------------------------------------------------------------
#define B_DIM 2
#define C_DIM 256
#define H_DIM 256
#define W_DIM 256
#define HW    (H_DIM * W_DIM)   // 65536

typedef __attribute__((ext_vector_type(2))) float v2f;
typedef __attribute__((ext_vector_type(8))) float v8f;

__device__ __forceinline__ v8f wmma_f32_k4(v2f a, v2f b, v8f c) {
    // D = A(16x4, f32) * B(4x16, f32) + C(16x16, f32)
    return __builtin_amdgcn_wmma_f32_16x16x4_f32(
        /*neg_a=*/false, a, /*neg_b=*/false, b,
        /*c_mod=*/(short)0, c, /*reuse_a=*/false, /*reuse_b=*/false);
}

// --------------------------- async global->LDS -----------------------------
#if defined(__has_builtin)
#  if __has_builtin(__builtin_amdgcn_global_load_async_to_lds_b32)
#    define HAVE_ASYNC_LDS 1
#  endif
#endif
#ifndef HAVE_ASYNC_LDS
#  define HAVE_ASYNC_LDS 0
#endif

typedef __attribute__((address_space(3))) int lds_int_t;
typedef __attribute__((address_space(1))) int glb_int_t;

__device__ __forceinline__ lds_int_t* as_lds(void* p) {
    // generic LDS pointer's low 32 bits are the LDS offset
    return (lds_int_t*)(uint32_t)(uintptr_t)p;
}
__device__ __forceinline__ glb_int_t* as_glb(const void* p) {
    return (glb_int_t*)(uintptr_t)p;
}
__device__ __forceinline__ void wait_async_zero() {
#if defined(__has_builtin)
#  if __has_builtin(__builtin_amdgcn_s_wait_asynccnt)
    __builtin_amdgcn_s_wait_asynccnt(0);
    return;
#  endif
#endif
#if HAVE_ASYNC_LDS
    asm volatile("s_wait_asynccnt 0" ::: "memory");
#endif
}

// ---------------------------------------------------------------------------
// Kernel 0: transpose proj_w  OIHW (co,ci,3,3) -> wT[tap][ci][co]
// ---------------------------------------------------------------------------
__global__ __launch_bounds__(256) void wtrans_kernel(const float* __restrict__ w,
                                                     float* __restrict__ wT) {
    int idx = blockIdx.x * 256 + threadIdx.x;     // 9*256*256 = 589824 total
    int co  = idx & 255;
    int ci  = (idx >> 8) & 255;
    int tap = idx >> 16;                          // 0..8
    wT[(tap << 16) + (ci << 8) + co] = w[((co << 8) + ci) * 9 + tap];
}

// ---------------------------------------------------------------------------
// Kernel 1: 3x3 conv (pad=1) as implicit GEMM with V_WMMA_F32_16X16X4_F32.
// Output NHWC fp32 (+bias).  Block tile: 64 pixels (one image row segment)
// x 64 out-channels; 8 waves, each owns two adjacent 16x16 C tiles.
// blockIdx.z selects feat0/feat2.
// ---------------------------------------------------------------------------
__global__ __launch_bounds__(256) void conv3x3_wmma_kernel(
    const float* __restrict__ in0, const float* __restrict__ in1,
    const float* __restrict__ wT,  const float* __restrict__ bias,
    float* __restrict__ out0, float* __restrict__ out1) {

    __shared__ float As[32][68];   // [k within chunk][pixel]
    __shared__ float Bs[32][68];   // [k within chunk][out-channel]

    const float* __restrict__ in  = blockIdx.z ? in1  : in0;
    float* __restrict__       out = blockIdx.z ? out1 : out0;

    const int tid  = threadIdx.x;
    const int lane = tid & 31;
    const int wave = tid >> 5;

    const int p0  = blockIdx.x * 64;       // pixel tile start (64 consecutive x)
    const int bb  = p0 >> 16;              // batch
    const int rem = p0 & (HW - 1);
    const int y   = rem >> 8;              // row
    const int x0  = rem & 255;             // col start (multiple of 64)
    const int co0 = blockIdx.y * 64;       // out-channel tile start

    const int m0    = (wave & 3) * 16;
    const int n0    = (wave >> 2) * 32;
    const int lm    = lane & 15;
    const int khalf = (lane < 16) ? 0 : 2;

    v8f acc0 = {0.f,0.f,0.f,0.f,0.f,0.f,0.f,0.f};
    v8f acc1 = {0.f,0.f,0.f,0.f,0.f,0.f,0.f,0.f};

    for (int ky = 0; ky < 3; ++ky) {
        const int  yy    = y + ky - 1;
        const bool rowok = ((unsigned)yy < (unsigned)H_DIM);
        for (int kx = 0; kx < 3; ++kx) {
            const float* wtap = wT + ((ky * 3 + kx) << 16) + co0;
            for (int kc = 0; kc < 8; ++kc) {
                const int ci0 = kc * 32;
                __syncthreads();
                #pragma unroll
                for (int r = 0; r < 8; ++r) {
                    int idx = r * 256 + tid;       // 2048 elements each
                    int k   = idx >> 6;            // 0..31 (channel within chunk)
                    int m   = idx & 63;            // pixel / out-channel
                    int xx  = x0 + m + kx - 1;
                    float av = 0.f;
                    if (rowok && (unsigned)xx < (unsigned)W_DIM)
                        av = in[(bb * C_DIM + ci0 + k) * HW + yy * W_DIM + xx];
                    As[k][m] = av;
                    Bs[k][m] = wtap[(ci0 + k) << 8 | m];
                }
                // prefetch next K-chunk (speculative; dropped if invalid)
                if (kc < 7) {
                    int k = tid >> 6, m = tid & 63;
                    if (rowok)
                        __builtin_prefetch(&in[(bb * C_DIM + ci0 + 32 + k) * HW +
                                               yy * W_DIM + x0 + m], 0, 1);
                    __builtin_prefetch(&wtap[(ci0 + 32 + k) << 8 | m], 0, 1);
                }
                __syncthreads();
                #pragma unroll
                for (int kk = 0; kk < 32; kk += 4) {
                    const int kr = kk + khalf;
                    v2f a, b0, b1;
                    a.x  = As[kr][m0 + lm];        a.y  = As[kr + 1][m0 + lm];
                    b0.x = Bs[kr][n0 + lm];        b0.y = Bs[kr + 1][n0 + lm];
                    b1.x = Bs[kr][n0 + 16 + lm];   b1.y = Bs[kr + 1][n0 + 16 + lm];
                    acc0 = wmma_f32_k4(a, b0, acc0);
                    acc1 = wmma_f32_k4(a, b1, acc1);
                }
            }
        }
    }

    // C/D layout: VGPR e -> M = m0 + e (+8 for upper half-wave), N = n0 + (lane&15)
    const float bv0 = bias[co0 + n0 + lm];
    const float bv1 = bias[co0 + n0 + 16 + lm];
    const int   Mb  = m0 + ((lane < 16) ? 0 : 8);
    #pragma unroll
    for (int e = 0; e < 8; ++e) {
        const int pix = p0 + Mb + e;
        float* o = out + (size_t)pix * C_DIM + co0;
        o[n0 + lm]      = acc0[e] + bv0;
        o[n0 + 16 + lm] = acc1[e] + bv1;
    }
}

// ---------------------------------------------------------------------------
// Kernel 2: per (window, shift-config):
//   corr = (q @ k^T) * C^-0.5 + mask   (WMMA, 64x64, K=256, staged in LDS)
//   then _flow_plain (rows), _flow_plain (cols), _select+_flow_mid,
//   scattered directly through the splice/roll (+up2) permutation.
// ---------------------------------------------------------------------------
__device__ __forceinline__ int regionOf(int i, int j, int wi, int wj, int sh, int sw) {
    int rh = 0, rw = 0;
    if (sh && wi == 31) rh = (i < 4) ? 1 : 2;   // rolled rows 248..251 / 252..255
    if (sw && wj == 31) rw = (j < 4) ? 1 : 2;
    return rh * 3 + rw;
}

__global__ __launch_bounds__(256) void winflow_kernel(
    const float* __restrict__ f0, const float* __restrict__ f2,
    float* __restrict__ outbuf) {

    __shared__ float Qs[32][68];     // [channel within chunk][window pixel]
    __shared__ float Ks[32][68];
    __shared__ float CORR[64][68];   // corr[l][t] (scaled + masked)

    const int tid  = threadIdx.x;
    const int lane = tid & 31;
    const int wave = tid >> 5;

    const int win = blockIdx.x;          // 0..2047 = b*1024 + wi*32 + wj
    const int cfg = blockIdx.y;          // 0..3  (bit1 = shift-h, bit0 = shift-w)
    const int bb  = win >> 10;
    const int wi  = (win >> 5) & 31;
    const int wj  = win & 31;
    const int sh  = (cfg & 2) ? 4 : 0;
    const int sw  = (cfg & 1) ? 4 : 0;
    const int chh = cfg >> 1, cww = cfg & 1;

    const int m0    = (wave & 3) * 16;
    const int n0    = (wave >> 2) * 32;
    const int lm    = lane & 15;
    const int khalf = (lane < 16) ? 0 : 2;

    v8f acc0 = {0.f,0.f,0.f,0.f,0.f,0.f,0.f,0.f};
    v8f acc1 = {0.f,0.f,0.f,0.f,0.f,0.f,0.f,0.f};

    for (int kc = 0; kc < 8; ++kc) {
        const int ci0 = kc * 32;
        __syncthreads();
        #pragma unroll
        for (int r = 0; r < 8; ++r) {
            int idx = r * 256 + tid;
            int pix = idx >> 5;                  // 0..63
            int c   = idx & 31;                  // channel (coalesced)
            int yy  = ((wi << 3) + (pix >> 3) + sh) & 255;  // roll(-sh)
            int xx  = ((wj << 3) + (pix & 7) + sw) & 255;
            size_t base = ((size_t)((bb << 16) + (yy << 8) + xx)) * C_DIM + ci0 + c;
#if HAVE_ASYNC_LDS
            // async DMA into LDS (ASYNCcnt-tracked); all lanes valid, no predication
            __builtin_amdgcn_global_load_async_to_lds_b32(
                as_glb(f0 + base), as_lds(&Qs[c][pix]), 0, 0);
            __builtin_amdgcn_global_load_async_to_lds_b32(
                as_glb(f2 + base), as_lds(&Ks[c][pix]), 0, 0);
#else
            Qs[c][pix] = f0[base];
            Ks[c][pix] = f2[base];
#endif
        }
#if HAVE_ASYNC_LDS
        wait_async_zero();
#endif
        __syncthreads();
        #pragma unroll
        for (int kk = 0; kk < 32; kk += 4) {
            const int kr = kk + khalf;
            v2f a, b0, b1;
            a.x  = Qs[kr][m0 + lm];       a.y  = Qs[kr + 1][m0 + lm];
            b0.x = Ks[kr][n0 + lm];       b0.y = Ks[kr + 1][n0 + lm];
            b1.x = Ks[kr][n0 + 16 + lm];  b1.y = Ks[kr + 1][n0 + 16 + lm];
            acc0 = wmma_f32_k4(a, b0, acc0);
            acc1 = wmma_f32_k4(a, b1, acc1);
        }
    }

    // scale + shift mask, write corr to LDS
    const float scale = 0.0625f;                 // 256^-0.5
    const int   Mb    = m0 + ((lane < 16) ? 0 : 8);
    #pragma unroll
    for (int e = 0; e < 8; ++e) {
        int l  = Mb + e;
        int rl = regionOf(l >> 3, l & 7, wi, wj, sh, sw);
        int t0 = n0 + lm, t1 = t0 + 16;
        int r0 = regionOf(t0 >> 3, t0 & 7, wi, wj, sh, sw);
        int r1 = regionOf(t1 >> 3, t1 & 7, wi, wj, sh, sw);
        CORR[l][t0] = acc0[e] * scale + ((rl != r0) ? -10000.f : 0.f);
        CORR[l][t1] = acc1[e] * scale + ((rl != r1) ? -10000.f : 0.f);
    }
    __syncthreads();

    float* __restrict__ out1 = outbuf;                 // flow      (2,2,512,512)
    float* __restrict__ out2 = outbuf + 1048576;       // -flow
    float* __restrict__ out3 = outbuf + 2097152;       // up2(flow02)
    float* __restrict__ out4 = outbuf + 3145728;       // up2(flow20)

    // ---- _flow_plain on rows (fo02) and on columns (fo20) ------------------
    {
        const int m  = tid >> 4;                 // 0..15 central pixel
        const int u  = tid & 15;                 // 16 lanes cooperate
        const int mi = m >> 2, mj = m & 3;
        const int l  = (mi + 2) * 8 + (mj + 2);
        #pragma unroll
        for (int variant = 0; variant < 2; ++variant) {
            float mx = -3.0e38f, S = 0.f, Sx = 0.f, Sy = 0.f;
            #pragma unroll
            for (int rr = 0; rr < 4; ++rr) {
                int   t  = u + rr * 16;
                float cv = variant ? CORR[t][l] : CORR[l][t];
                float nm = fmaxf(mx, cv);
                float fs = expf(mx - nm);
                float e  = expf(cv - nm);
                S  = S * fs + e;
                Sx = Sx * fs + e * (float)(t & 7);
                Sy = Sy * fs + e * (float)(t >> 3);
                mx = nm;
            }
            #pragma unroll
            for (int off = 1; off < 16; off <<= 1) {
                float m2  = __shfl_xor(mx, off, 16);
                float S2  = __shfl_xor(S,  off, 16);
                float Sx2 = __shfl_xor(Sx, off, 16);
                float Sy2 = __shfl_xor(Sy, off, 16);
                float nm  = fmaxf(mx, m2);
                float a1  = expf(mx - nm), a2 = expf(m2 - nm);
                S  = S * a1 + S2 * a2;
                Sx = Sx * a1 + Sx2 * a2;
                Sy = Sy * a1 + Sy2 * a2;
                mx = nm;
            }
            if (u == 0) {
                float fx = Sx / S - (float)(mj + 2);
                float fy = Sy / S - (float)(mi + 2);
                // splice (factor 1): Yo = 8*wi + 4*cfg_h + mi + 2 (mod 256)
                int Yo = ((wi << 3) + (chh << 2) + mi + 2) & 255;
                int Xo = ((wj << 3) + (cww << 2) + mj + 2) & 255;
                float* dst = variant ? out4 : out3;
                #pragma unroll
                for (int f = 0; f < 2; ++f) {
                    float v = 2.f * (f ? fy : fx);        // up2 scale
                    size_t base = ((size_t)(bb * 2 + f) * 512 + 2 * Yo) * 512 + 2 * Xo;
                    dst[base] = v; dst[base + 1] = v;
                    dst[base + 512] = v; dst[base + 513] = v;
                }
            }
        }
    }

    // ---- _select + _flow_mid (flow, 8x8 per window) ------------------------
    {
        const int p  = tid >> 2;                 // output pixel 0..63
        const int u  = tid & 3;                  // 4 lanes: one neighbor offset each
        const int i  = p >> 3, j = p & 7;
        const int jh = i + (u >> 1);             // 0..8
        const int jw = j + (u & 1);
        float mx = -3.0e38f, S = 0.f, Sx = 0.f, Sy = 0.f;
        #pragma unroll
        for (int l = 0; l < 64; ++l) {
            int h0 = l >> 3, w0 = l & 7;
            int ih = jh + 3 - h0, iw = jw + 3 - w0;
            float cv = 0.f;                       // invalid entries are literal 0
            if ((unsigned)ih < 8u && (unsigned)iw < 8u) cv = CORR[l][ih * 8 + iw];
            float frx = 2.f * (float)w0 - (float)jw - 3.f;   // 2*(w0 - (jw+3)/2)
            float fry = 2.f * (float)h0 - (float)jh - 3.f;
            float nm = fmaxf(mx, cv);
            float fs = expf(mx - nm);
            float e  = expf(cv - nm);
            S  = S * fs + e;
            Sx = Sx * fs + e * frx;
            Sy = Sy * fs + e * fry;
            mx = nm;
        }
        #pragma unroll
        for (int off = 1; off < 4; off <<= 1) {
            float m2  = __shfl_xor(mx, off, 4);
            float S2  = __shfl_xor(S,  off, 4);
            float Sx2 = __shfl_xor(Sx, off, 4);
            float Sy2 = __shfl_xor(Sy, off, 4);
            float nm  = fmaxf(mx, m2);
            float a1  = expf(mx - nm), a2 = expf(m2 - nm);
            S  = S * a1 + S2 * a2;
            Sx = Sx * a1 + Sx2 * a2;
            Sy = Sy * a1 + Sy2 * a2;
            mx = nm;
        }
        if (u == 0) {
            float fx = Sx / S, fy = Sy / S;
            // splice (factor 2): Y = 16*wi + 8*cfg_h + i + 4 (mod 512)
            int Y = ((wi << 4) + (chh << 3) + i + 4) & 511;
            int X = ((wj << 4) + (cww << 3) + j + 4) & 511;
            size_t b0 = ((size_t)(bb * 2 + 0) * 512 + Y) * 512 + X;
            size_t b1 = ((size_t)(bb * 2 + 1) * 512 + Y) * 512 + X;
            out1[b0] =  fx;  out1[b1] =  fy;
            out2[b0] = -fx;  out2[b1] = -fy;
        }
    }
}

// ---------------------------------------------------------------------------
// Kernel 3: range0/range1 outputs are data-independent closed forms of the
// output coordinate (splice+roll inverted analytically).
// ---------------------------------------------------------------------------
__global__ __launch_bounds__(256) void ranges_kernel(float* __restrict__ outbuf) {
    int idx = blockIdx.x * 256 + threadIdx.x;    // over 2*4*512*512
    float* out5 = outbuf + 4194304;              // up2(range0)
    float* out6 = outbuf + 6291456;              // range1
    int X2 = idx & 511;
    int Y2 = (idx >> 9) & 511;
    int c4 = (idx >> 18) & 3;

    // range0: r0 = [-(mj+2), -(mi+2), 5-mj, 5-mi], up2 multiplies by 2
    int mi = ((Y2 >> 1) + 2) & 3;
    int mj = ((X2 >> 1) + 2) & 3;
    float v5 = (c4 == 0) ? -2.f * (float)(mj + 2)
             : (c4 == 1) ? -2.f * (float)(mi + 2)
             : (c4 == 2) ?  2.f * (float)(5 - mj)
                         :  2.f * (float)(5 - mi);
    out5[idx] = v5;

    // range1: rng_f(n) = min((n+3)/2, 7-(n+3)/2);  r = min over 2x2 neighborhood
    int i = (Y2 + 4) & 7;
    int j = (X2 + 4) & 7;
    float bj0 = 0.5f * (float)(j + 3), bj1 = 0.5f * (float)(j + 4);
    float bi0 = 0.5f * (float)(i + 3), bi1 = 0.5f * (float)(i + 4);
    float rx = fminf(fminf(bj0, 7.f - bj0), fminf(bj1, 7.f - bj1));
    float ry = fminf(fminf(bi0, 7.f - bi0), fminf(bi1, 7.f - bi1));
    float v6 = (c4 == 0) ? -2.f * rx
             : (c4 == 1) ? -2.f * ry
             : (c4 == 2) ?  2.f * rx
                         :  2.f * ry;
    out6[idx] = v6;
}

// ---------------------------------------------------------------------------
// Host launcher
// ---------------------------------------------------------------------------
extern "C" void kernel_launch(void* const* d_in, const int* in_sizes, int n_in,
                              void* d_out, int out_size, void* d_ws, size_t ws_size,
                              hipStream_t stream) {
    (void)in_sizes; (void)n_in; (void)out_size; (void)ws_size;
    const float* feat0  = (const float*)d_in[0];
    const float* feat1  = (const float*)d_in[1];
    // d_in[2], d_in[3] (score0/score2) are unused by the reference forward.
    const float* proj_w = (const float*)d_in[4];
    const float* proj_b = (const float*)d_in[5];
    float* out = (float*)d_out;

    float* ws = (float*)d_ws;
    float* wT = ws;                              // 9*256*256       = 589824 f
    float* f0 = ws + 589824;                     // B*H*W*C (NHWC)  = 33554432 f
    float* f2 = f0 + (size_t)B_DIM * HW * C_DIM; // another 33554432 f

    // 0) weight layout transform
    wtrans_kernel<<<2304, 256, 0, stream>>>(proj_w, wT);

    // 1) conv3x3 for feat0 and feat2 (z selects input)
    dim3 gconv((B_DIM * HW) / 64, C_DIM / 64, 2); // 2048 x 4 x 2
    conv3x3_wmma_kernel<<<gconv, 256, 0, stream>>>(feat0, feat1, wT, proj_b, f0, f2);

    // 2) windows x shift-configs: corr GEMM + all flows + scatter
    dim3 gwin(B_DIM * 32 * 32, 4);                // 2048 x 4
    winflow_kernel<<<gwin, 256, 0, stream>>>(f0, f2, out);

    // 3) analytic range outputs
    ranges_kernel<<<(2 * 4 * 512 * 512) / 256, 256, 0, stream>>>(out);
}